// PEER_18622978195831
// MI455X (gfx1250) — compile-verified
//
#include <hip/hip_runtime.h>
#include <math.h>

typedef float v2f __attribute__((ext_vector_type(2)));
typedef float v8f __attribute__((ext_vector_type(8)));

#define DIM        512
#define HEADS      8
#define DIM_KEY    32
#define NUM_KEYS   256
#define TOPK       16
#define NTOK       1024          // b * n = 2 * 512
#define BN_SCALE   0.99999500003749978f   // 1/sqrt(1 + 1e-5)

// ---------------------------------------------------------------------------
// Kernel 1: q = (x @ w_q) * BN_SCALE
// C[1024,512] = A[1024,512] * B[512,512], fp32 WMMA 16x16x4, 1 tile per wave.
// A frag (16x4): lane L holds M = L%16; vgpr v holds K = k0 + 2*(L/16) + v.
// C/D (16x16):   vgpr r -> M = r + 8*(L/16), N = L%16.
// ---------------------------------------------------------------------------
__global__ void peer_gemm_q(const float* __restrict__ A,
                            const float* __restrict__ B,
                            float* __restrict__ Q) {
    const int wave = (blockIdx.x * blockDim.x + threadIdx.x) >> 5;
    const int lane = threadIdx.x & 31;
    const int tn   = wave & 31;        // 512/16 = 32 N tiles
    const int tm   = wave >> 5;        // 64 M tiles
    const int half = lane >> 4;
    const int l16  = lane & 15;

    const int row = tm * 16 + l16;     // A row for this lane
    const int col = tn * 16 + l16;     // B col for this lane

    v8f c = {};
    const float* arow = A + row * DIM;
    for (int k0 = 0; k0 < DIM; k0 += 4) {
        const int ka = k0 + 2 * half;
        v2f a, b;
        a.x = arow[ka];
        a.y = arow[ka + 1];
        b.x = B[ka * DIM + col];
        b.y = B[(ka + 1) * DIM + col];
        c = __builtin_amdgcn_wmma_f32_16x16x4_f32(false, a, false, b,
                                                  (short)0, c, false, false);
    }
#pragma unroll
    for (int r = 0; r < 8; ++r) {
        const int m = tm * 16 + r + 8 * half;
        Q[m * DIM + tn * 16 + l16] = c[r] * BN_SCALE;
    }
}

// ---------------------------------------------------------------------------
// Kernel 2: sim0[h][t][k] = sum_d q[t][h*32+d] * keys[h][k][0][d]
// Only the p=0 plane is needed (p=1 path is dead in the reference).
// Per head: [1024 x 32] @ [32 x 256] with fp32 WMMA, K-loop of 8.
// ---------------------------------------------------------------------------
__global__ void peer_sim(const float* __restrict__ Q,
                         const float* __restrict__ keys,
                         float* __restrict__ S) {
    int w = (blockIdx.x * blockDim.x + threadIdx.x) >> 5;
    const int lane = threadIdx.x & 31;
    const int tn = w & 15;   w >>= 4;  // 256/16 = 16 N tiles
    const int tm = w & 63;   w >>= 6;  // 64 M tiles
    const int h  = w;                  // 8 heads
    const int half = lane >> 4;
    const int l16  = lane & 15;

    const int row  = tm * 16 + l16;
    const int kcol = tn * 16 + l16;
    const float* arow = Q + row * DIM + h * DIM_KEY;
    const float* brow = keys + ((h * NUM_KEYS + kcol) * 2 + 0) * DIM_KEY;

    v8f c = {};
    for (int k0 = 0; k0 < DIM_KEY; k0 += 4) {
        const int d = k0 + 2 * half;
        v2f a, b;
        a.x = arow[d];
        a.y = arow[d + 1];
        b.x = brow[d];
        b.y = brow[d + 1];
        c = __builtin_amdgcn_wmma_f32_16x16x4_f32(false, a, false, b,
                                                  (short)0, c, false, false);
    }
#pragma unroll
    for (int r = 0; r < 8; ++r) {
        const int m = tm * 16 + r + 8 * half;
        S[(h * NTOK + m) * NUM_KEYS + tn * 16 + l16] = c[r];
    }
}

// ---------------------------------------------------------------------------
// Kernel 3: per (head, token) row of 256 sims:
//   top-16 (desc, stable: larger value wins, tie -> smaller index)  -> v0,i0
//   combos  all[i*16+j] = v0[i] + float(i0[j]); top-16 of those      -> sc,si
//   gates = softmax(sc); emit pk (flat positions) + gates.
// One wave32 per row; values live 8-per-lane; argmax via shfl_xor tree.
// ---------------------------------------------------------------------------
__global__ void peer_topk(const float* __restrict__ S,
                          int* __restrict__ pk,
                          float* __restrict__ gate) {
    const int wg   = (blockIdx.x * blockDim.x + threadIdx.x) >> 5;
    const int lane = threadIdx.x & 31;
    const int h    = wg >> 10;         // / 1024
    const int tok  = wg & 1023;

    const float* row = S + (h * NTOK + tok) * NUM_KEYS;
    float v[8];
#pragma unroll
    for (int j = 0; j < 8; ++j) v[j] = row[lane * 8 + j];

    float v0[TOPK]; int i0[TOPK];
#pragma unroll 1
    for (int it = 0; it < TOPK; ++it) {
        float bv = v[0]; int bi = lane * 8;
#pragma unroll
        for (int j = 1; j < 8; ++j)
            if (v[j] > bv) { bv = v[j]; bi = lane * 8 + j; }
        for (int m = 16; m >= 1; m >>= 1) {
            const float ov = __shfl_xor(bv, m, 32);
            const int   oi = __shfl_xor(bi, m, 32);
            if (ov > bv || (ov == bv && oi < bi)) { bv = ov; bi = oi; }
        }
        v0[it] = bv; i0[it] = bi;
        if ((bi >> 3) == lane) v[bi & 7] = -__builtin_inff();
    }

    // combos: faithful to the source's tuple-unpack bug: values0 + indices0
#pragma unroll
    for (int j = 0; j < 8; ++j) {
        const int flat = lane * 8 + j;
        v[j] = v0[flat >> 4] + (float)i0[flat & 15];
    }

    float sc[TOPK]; int si[TOPK];
#pragma unroll 1
    for (int it = 0; it < TOPK; ++it) {
        float bv = v[0]; int bi = lane * 8;
#pragma unroll
        for (int j = 1; j < 8; ++j)
            if (v[j] > bv) { bv = v[j]; bi = lane * 8 + j; }
        for (int m = 16; m >= 1; m >>= 1) {
            const float ov = __shfl_xor(bv, m, 32);
            const int   oi = __shfl_xor(bi, m, 32);
            if (ov > bv || (ov == bv && oi < bi)) { bv = ov; bi = oi; }
        }
        sc[it] = bv; si[it] = bi;
        if ((bi >> 3) == lane) v[bi & 7] = -__builtin_inff();
    }

    // softmax over the 16 selected scores (sc[0] is the max: sorted desc)
    float e[TOPK]; float sum = 0.f;
#pragma unroll
    for (int k = 0; k < TOPK; ++k) { e[k] = __expf(sc[k] - sc[0]); sum += e[k]; }
    const float inv = 1.f / sum;

    if (lane == 0) {
        const int base = (tok * HEADS + h) * TOPK;
#pragma unroll
        for (int k = 0; k < TOPK; ++k) {
            pk[base + k]   = si[k];
            gate[base + k] = e[k] * inv;
        }
    }
}

// ---------------------------------------------------------------------------
// Kernel 4: per token (256-thread block):
//   hval[e] = gelu( dot(q, Wd[pk[e]]) ) * gate[e]   for e = h*16+k in [0,128)
//   out[d]  = sum_e hval[e] * Wu[pk[e]][d]
// q staged in LDS; 2 lanes per expert for the 512-dot (shfl-combine);
// up-accumulation reads are contiguous across lanes (L2-hot rows 0..255).
// ---------------------------------------------------------------------------
__global__ void peer_experts(const float* __restrict__ Q,
                             const int* __restrict__ pk,
                             const float* __restrict__ gate,
                             const float* __restrict__ Wd,
                             const float* __restrict__ Wu,
                             float* __restrict__ out) {
    __shared__ float q_lds[DIM];
    __shared__ float hval[HEADS * TOPK];
    __shared__ int   eidx[HEADS * TOPK];

    const int t   = blockIdx.x;     // token
    const int tid = threadIdx.x;    // 0..255

    q_lds[tid]       = Q[t * DIM + tid];
    q_lds[tid + 256] = Q[t * DIM + tid + 256];
    if (tid < HEADS * TOPK) eidx[tid] = pk[t * (HEADS * TOPK) + tid];
    __syncthreads();

    {
        const int e  = tid >> 1;
        const int hf = tid & 1;
        const float* wrow = Wd + (size_t)eidx[e] * DIM + hf * 256;
        const float* qh   = q_lds + hf * 256;
        float acc = 0.f;
        for (int d = 0; d < 256; ++d) acc += qh[d] * wrow[d];
        acc += __shfl_xor(acc, 1, 32);          // pair lanes share a wave
        if (hf == 0) {
            const float x  = acc;
            const float x3 = x * x * x;
            const float g  = 0.5f * x *
                (1.f + tanhf(0.7978845608028654f * (x + 0.044715f * x3)));
            hval[e] = g * gate[t * (HEADS * TOPK) + e];
        }
    }
    __syncthreads();

    float o0 = 0.f, o1 = 0.f;
    for (int e = 0; e < HEADS * TOPK; ++e) {
        const float* urow = Wu + (size_t)eidx[e] * DIM;
        const float hv = hval[e];
        o0 += hv * urow[tid];
        o1 += hv * urow[tid + 256];
    }
    out[t * DIM + tid]       = o0;
    out[t * DIM + tid + 256] = o1;
}

// ---------------------------------------------------------------------------
// Launcher. Workspace layout (floats): q[1024*512] | sim[8*1024*256] |
// pk[8192*16] (int) | gate[8192*16]  -> ~11 MB total.
// ---------------------------------------------------------------------------
extern "C" void kernel_launch(void* const* d_in, const int* in_sizes, int n_in,
                              void* d_out, int out_size, void* d_ws, size_t ws_size,
                              hipStream_t stream) {
    (void)in_sizes; (void)n_in; (void)out_size; (void)ws_size;

    const float* x    = (const float*)d_in[0];   // [2,512,512]
    const float* w_q  = (const float*)d_in[1];   // [512,512]
    const float* keys = (const float*)d_in[2];   // [8,256,2,32]
    const float* Wd   = (const float*)d_in[3];   // [65536,512]
    const float* Wu   = (const float*)d_in[4];   // [65536,512]
    float* out = (float*)d_out;                  // [2,512,512]

    float* q    = (float*)d_ws;                        // 1024*512
    float* sim  = q + NTOK * DIM;                      // 8*1024*256
    int*   pk   = (int*)(sim + HEADS * NTOK * NUM_KEYS);
    float* gate = (float*)(pk + NTOK * HEADS * TOPK);

    // 1) q GEMM: 64x32 = 2048 wave-tiles, 8 waves/block
    peer_gemm_q<<<256, 256, 0, stream>>>(x, w_q, q);
    // 2) sim (p=0 only): 8*64*16 = 8192 wave-tiles
    peer_sim<<<1024, 256, 0, stream>>>(q, keys, sim);
    // 3) double top-16 + softmax gates: 8192 wave-rows
    peer_topk<<<1024, 256, 0, stream>>>(sim, pk, gate);
    // 4) expert gather / GELU / gated up-projection: 1 block per token
    peer_experts<<<NTOK, 256, 0, stream>>>(q, pk, gate, Wd, Wu, out);
}